// DC_and_CE_and_self_loss_65085934404178
// MI455X (gfx1250) — compile-verified
//
#include <hip/hip_runtime.h>

// Problem dimensions (fixed by reference setup_inputs)
#define BB   2
#define CC   32
#define DD   64
#define HH   128
#define WW   128
#define NCLS 2
#define SPAT (DD*HH*WW)      // 1048576
#define NVOX (BB*SPAT)       // 2097152
#define TOPN 250u

// Workspace layout (byte offsets)
#define OFF_ACC   0              // 256 floats: scalar accumulators + std + std_n
#define OFF_RS    4096           // 512 uints: radix histograms + select state
#define OFF_PART  (64*1024)      // partial-sum region (<= 32KB used)
#define OFF_SIM   (2*1024*1024)  // 2,097,152 floats (8MB)
#define OFF_POSM  (10*1024*1024) // pos mask, u8
#define OFF_MA    (12*1024*1024) // scratch mask A
#define OFF_MB    (14*1024*1024) // scratch mask B

typedef float    v2f  __attribute__((ext_vector_type(2)));
typedef float    v8f  __attribute__((ext_vector_type(8)));
typedef unsigned u32x4 __attribute__((ext_vector_type(4)));
typedef unsigned u32x8 __attribute__((ext_vector_type(8)));

__device__ inline float blockReduceSum(float v, float* smem) {
  int t = threadIdx.x;
  smem[t] = v; __syncthreads();
  for (int s = blockDim.x >> 1; s > 0; s >>= 1) {
    if (t < s) smem[t] += smem[t + s];
    __syncthreads();
  }
  float r = smem[0]; __syncthreads();
  return r;
}

__global__ void init_ws(float* acc, unsigned* rs) {
  int t = threadIdx.x;
  if (t < 256) acc[t] = 0.f;
  if (t < 256) rs[t] = 0u;
  if (t == 0) { rs[256] = 0u; rs[257] = TOPN; rs[258] = 0u; rs[259] = 0u; }
}

// CE + dice partials; also writes the positive mask.
__global__ void ce_dice_partial(const float* __restrict__ nout, const int* __restrict__ tgt,
                                unsigned char* __restrict__ posm, float* __restrict__ part) {
  float ce = 0.f, tp = 0.f, sp = 0.f, cn = 0.f;
  int stride = gridDim.x * blockDim.x;
  for (int v = blockIdx.x * blockDim.x + threadIdx.x; v < NVOX; v += stride) {
    int b = v / SPAT, s = v - b * SPAT;
    float x0 = nout[(size_t)(b * NCLS + 0) * SPAT + s];
    float x1 = nout[(size_t)(b * NCLS + 1) * SPAT + s];
    int t = tgt[v];
    float mx = fmaxf(x0, x1);
    float lse = mx + logf(expf(x0 - mx) + expf(x1 - mx));
    ce += (t ? x1 : x0) - lse;           // sum of log p(target); negate at end
    float p1 = expf(x1 - lse);
    sp += p1;
    if (t == 1) { tp += p1; cn += 1.f; }
    posm[v] = (unsigned char)(t == 1);
  }
  __shared__ float sm[256];
  float r;
  r = blockReduceSum(ce, sm); if (!threadIdx.x) part[blockIdx.x * 4 + 0] = r;
  r = blockReduceSum(tp, sm); if (!threadIdx.x) part[blockIdx.x * 4 + 1] = r;
  r = blockReduceSum(sp, sm); if (!threadIdx.x) part[blockIdx.x * 4 + 2] = r;
  r = blockReduceSum(cn, sm); if (!threadIdx.x) part[blockIdx.x * 4 + 3] = r;
}

// Deterministic final reduction: column-wise sum of part[nrows][ncols] -> dst[ncols]
__global__ void reduce_cols(const float* __restrict__ part, int nrows, int ncols,
                            float* __restrict__ dst) {
  __shared__ float sm[256];
  for (int c = 0; c < ncols; ++c) {
    float a = 0.f;
    for (int r = threadIdx.x; r < nrows; r += blockDim.x) a += part[(size_t)r * ncols + c];
    float s = blockReduceSum(a, sm);
    if (!threadIdx.x) dst[c] = s;
    __syncthreads();
  }
}

// Masked per-channel sum of feature over positive voxels.
// The 8KB mask chunk is staged into LDS once via the Tensor Data Mover
// (1-D D# descriptor, TENSORcnt-tracked) and re-read 32x (once per channel).
#define STD_CHUNK 8192
__global__ void std_partial(const float* __restrict__ feat, const unsigned char* __restrict__ posm,
                            float* __restrict__ part) {
  __shared__ __attribute__((aligned(16))) unsigned char msk[STD_CHUNK];
  __shared__ float sm[256];
  int base = blockIdx.x * STD_CHUNK;
  int b = base / SPAT;
  int s0 = base - b * SPAT;

  if (threadIdx.x < 32) {   // one wave issues the TDM op (TDM ignores EXEC; wave-level issue)
    unsigned long long ga = (unsigned long long)(size_t)(posm + base);
    unsigned ldsAddr = (unsigned)(size_t)&msk[0];   // flat LDS aperture: offset in addr[31:0]
    u32x4 g0;
    g0[0] = 1u;                                   // count=1 valid descriptor, user mode
    g0[1] = ldsAddr;                              // lds_addr (bytes)
    g0[2] = (unsigned)(ga & 0xffffffffu);         // global_addr[31:0]
    g0[3] = (unsigned)(ga >> 32) | 0x80000000u;   // global_addr[56:32] | type=2 @ [127:126]
    u32x8 g1;
    g1[0] = 0u;                                   // data_size=1B, no multicast/pad/flags
    g1[1] = ((unsigned)STD_CHUNK & 0xffffu) << 16;           // tensor_dim0[15:0] @ [63:48]
    g1[2] = ((unsigned)STD_CHUNK >> 16) | (1u << 16);        // tensor_dim0[31:16], tensor_dim1=1
    g1[3] = ((unsigned)STD_CHUNK & 0xffffu) << 16;           // tile_dim0 @ [127:112]
    g1[4] = 1u;                                   // tile_dim1=1, tile_dim2=0
    g1[5] = (unsigned)STD_CHUNK;                  // tensor_dim0_stride[31:0]
    g1[6] = ((unsigned)STD_CHUNK & 0xffffu) << 16;           // tensor_dim1_stride[15:0] @ [223:208]
    g1[7] = 0u;
    asm volatile("tensor_load_to_lds %0, %1\n\t"
                 "s_wait_tensorcnt 0x0"
                 :: "s"(g0), "s"(g1) : "memory");
  }
  __syncthreads();

  for (int c = 0; c < CC; ++c) {
    const float* fp = feat + (size_t)(b * CC + c) * SPAT + s0;
    float a = 0.f;
    for (int i = threadIdx.x; i < STD_CHUNK; i += blockDim.x)
      if (msk[i]) a += fp[i];
    float r = blockReduceSum(a, sm);
    if (!threadIdx.x) part[blockIdx.x * CC + c] = r;
  }
}

// std = sum/cnt (0 if cnt==0); std_n = std / max(||std||, 1e-12) -> acc[64..95]
__global__ void finalize_std(float* acc) {
  __shared__ float st[32];
  int t = threadIdx.x;
  float cnt = acc[3];
  float v = 0.f;
  if (t < 32) {
    v = (cnt > 0.f) ? acc[16 + t] / fmaxf(cnt, 1.f) : 0.f;
    st[t] = v * v;
  }
  __syncthreads();
  if (t == 0) {
    float n = 0.f;
    for (int i = 0; i < 32; ++i) n += st[i];
    st[0] = fmaxf(sqrtf(n), 1e-12f);
  }
  __syncthreads();
  if (t < 32) acc[64 + t] = v / st[0];
}

// sim = (feat . std_n) / max(||feat||, 1e-12) per voxel, GEMV via V_WMMA_F32_16X16X4_F32.
// Each wave: 16 voxels per tile; A = 16x4 feature chunk, B columns all = std_n chunk.
#define SIM_WAVES 8
#define SIM_TILES 8
__global__ void sim_wmma(const float* __restrict__ feat, const unsigned char* __restrict__ posm,
                         const float* __restrict__ acc, float* __restrict__ sim,
                         float* __restrict__ part) {
  int lane = threadIdx.x & 31;
  int wave = threadIdx.x >> 5;
  int m = lane & 15, half = lane >> 4;

  float bs0[8], bs1[8];
#pragma unroll
  for (int kc = 0; kc < 8; ++kc) {          // B layout mirrors A: (vgpr j, half h) -> K = 2h + j
    bs0[kc] = acc[64 + 4 * kc + 2 * half];
    bs1[kc] = acc[64 + 4 * kc + 2 * half + 1];
  }

  size_t tile0 = ((size_t)blockIdx.x * SIM_WAVES + wave) * SIM_TILES;
  float pacc = 0.f;
  for (int tt = 0; tt < SIM_TILES; ++tt) {
    size_t vbase = (tile0 + (size_t)tt) * 16;
    int b = (int)(vbase / SPAT);
    size_t s = vbase - (size_t)b * SPAT;
    v8f cacc = {0.f, 0.f, 0.f, 0.f, 0.f, 0.f, 0.f, 0.f};
    float q = 0.f;
#pragma unroll
    for (int kc = 0; kc < 8; ++kc) {
      int ch0 = 4 * kc + 2 * half;
      const float* p0 = feat + (size_t)(b * CC + ch0) * SPAT + s + (size_t)m;
      float a0 = p0[0];
      float a1 = p0[SPAT];                  // next channel
      q += a0 * a0 + a1 * a1;
      v2f A  = {a0, a1};
      v2f Bv = {bs0[kc], bs1[kc]};
      cacc = __builtin_amdgcn_wmma_f32_16x16x4_f32(
          false, A, false, Bv, (short)0, cacc, false, false);
    }
    float qt = q + __shfl_xor(q, 16, 32);   // combine the two channel halves (wave32)
    // D layout: lanes 0-15 hold M=vgpr (N=lane); lanes 16-31 hold M=8+vgpr (N=lane-16).
    bool writer = (lane < 8) || (lane >= 24);
    int idx = (lane < 16) ? (lane & 7) : ((lane - 16) & 7);
    float sdot = cacc[0];
#pragma unroll
    for (int i = 1; i < 8; ++i) sdot = (idx == i) ? cacc[i] : sdot;
    if (writer) {
      float sv = sdot / fmaxf(sqrtf(qt), 1e-12f);
      size_t vi = vbase + (size_t)m;
      sim[vi] = sv;
      if (posm[vi]) pacc += (1.f - sv);
    }
  }
  __shared__ float sm[256];
  float r = blockReduceSum(pacc, sm);
  if (!threadIdx.x) part[blockIdx.x] = r;
}

// W-axis binary max filter (radius 10, clamped): one block per 128-voxel row.
// Row is staged into LDS with CDNA5 async global->LDS copies (ASYNCcnt-tracked).
__global__ void dilate1d_w(const unsigned char* __restrict__ in, unsigned char* __restrict__ out) {
  __shared__ __attribute__((aligned(16))) unsigned char row[WW];
  int rbase = blockIdx.x * WW;
  if (threadIdx.x < 32) {                 // wave 0 stages 128B = 32 dwords, one per lane
    unsigned ldsAddr = (unsigned)(size_t)&row[threadIdx.x * 4];
    const unsigned char* g = in + rbase + threadIdx.x * 4;
    asm volatile("global_load_async_to_lds_b32 %0, %1, off"
                 :: "v"(ldsAddr), "v"(g) : "memory");
    asm volatile("s_wait_asynccnt 0x0" ::: "memory");
  }
  __syncthreads();
  int w = threadIdx.x;                    // blockDim.x == WW
  int lo = w - 10; if (lo < 0) lo = 0;
  int hi = w + 10; if (hi > WW - 1) hi = WW - 1;
  unsigned char m = 0;
  for (int i = lo; i <= hi; ++i) m |= row[i];
  out[rbase + w] = m;
}

// Generic strided 1-D binary max filter, radius 10, clamped borders (axis 1=H, 2=D).
__global__ void dilate1d(const unsigned char* __restrict__ in, unsigned char* __restrict__ out,
                         int axis) {
  int v = blockIdx.x * blockDim.x + threadIdx.x;
  if (v >= NVOX) return;
  int b = v / SPAT, s = v - b * SPAT;
  int d = s / (HH * WW), r = s - d * HH * WW;
  int h = r / WW;
  int coord, len, stride;
  if (axis == 1) { coord = h; len = HH; stride = WW; }
  else           { coord = d; len = DD; stride = HH * WW; }
  int lo = coord - 10; if (lo < 0) lo = 0;
  int hi = coord + 10; if (hi > len - 1) hi = len - 1;
  const unsigned char* p = in + (size_t)v + (size_t)(lo - coord) * stride;
  unsigned char m = 0;
  for (int i = lo; i <= hi; ++i) { m |= *p; p += stride; }
  out[v] = m;
}

// sum(relu(sim)) and count over (dil && !pos)
__global__ void masked_relu_reduce(const float* __restrict__ sim, const unsigned char* __restrict__ dil,
                                   const unsigned char* __restrict__ posm, float* __restrict__ part) {
  float s = 0.f, c = 0.f;
  int stride = gridDim.x * blockDim.x;
  for (int v = blockIdx.x * blockDim.x + threadIdx.x; v < NVOX; v += stride) {
    if (dil[v] && !posm[v]) {
      float x = sim[v];
      s += (x > 0.f) ? x : 0.f;
      c += 1.f;
    }
  }
  __shared__ float sm[256];
  float r;
  r = blockReduceSum(s, sm); if (!threadIdx.x) part[blockIdx.x * 2 + 0] = r;
  r = blockReduceSum(c, sm); if (!threadIdx.x) part[blockIdx.x * 2 + 1] = r;
}

__device__ inline unsigned keyOf(float f) {   // monotonic float->uint (descending-safe)
  unsigned u = __float_as_uint(f);
  return (u & 0x80000000u) ? ~u : (u | 0x80000000u);
}

// 8-bit radix histogram of neg-voxel keys matching the current prefix (integer atomics only)
__global__ void topk_hist(const float* __restrict__ sim, const unsigned char* __restrict__ posm,
                          unsigned* __restrict__ rs, int level) {
  unsigned prefix = rs[256];
  __shared__ unsigned lh[256];
  for (int i = threadIdx.x; i < 256; i += blockDim.x) lh[i] = 0u;
  __syncthreads();
  int stride = gridDim.x * blockDim.x;
  int shift = 24 - 8 * level;
  for (int v = blockIdx.x * blockDim.x + threadIdx.x; v < NVOX; v += stride) {
    if (!posm[v]) {
      unsigned k = keyOf(sim[v]);
      bool match = (level == 0) || ((k >> (shift + 8)) == prefix);
      if (match) atomicAdd(&lh[(k >> shift) & 255u], 1u);
    }
  }
  __syncthreads();
  for (int i = threadIdx.x; i < 256; i += blockDim.x)
    if (lh[i]) atomicAdd(&rs[i], lh[i]);
}

// Pick the bin containing the K-th largest; refine prefix; reset histogram.
__global__ void topk_select(unsigned* rs, int level) {
  if (threadIdx.x != 0 || blockIdx.x != 0) return;
  unsigned K = rs[257];
  unsigned cum = 0; int sel = 0;
  for (int b = 255; b >= 0; --b) {
    unsigned c = rs[b];
    if (cum + c >= K) { sel = b; break; }
    cum += c;
  }
  rs[256] = (rs[256] << 8) | (unsigned)sel;
  rs[257] = (K > cum) ? (K - cum) : 1u;     // remaining quota within equal-key bin
  for (int i = 0; i < 256; ++i) rs[i] = 0u; // clear hist for next level
}

// hsm: key>T -> 1; key==T -> first `need` tickets; else 0 (ties measure-zero in practice)
__global__ void hsm_build(const float* __restrict__ sim, const unsigned char* __restrict__ posm,
                          unsigned* __restrict__ rs, unsigned char* __restrict__ hsm) {
  unsigned T = rs[256], need = rs[257];
  int stride = gridDim.x * blockDim.x;
  for (int v = blockIdx.x * blockDim.x + threadIdx.x; v < NVOX; v += stride) {
    unsigned char o = 0;
    if (!posm[v]) {
      unsigned k = keyOf(sim[v]);
      if (k > T) o = 1;
      else if (k == T) { if (atomicAdd(&rs[259], 1u) < need) o = 1; }
    }
    hsm[v] = o;
  }
}

__global__ void final_combine(const float* __restrict__ acc, float* __restrict__ out) {
  if (threadIdx.x != 0 || blockIdx.x != 0) return;
  float N = (float)NVOX;
  float ce   = -acc[0] / N;
  float tp1  = acc[1], sp1 = acc[2], cnt = acc[3];
  float dc1  = (2.f * tp1 + 1e-5f) / (sp1 + cnt + 1e-5f + 1e-8f); // 2tp+fp+fn = sumP1+cnt1
  float dice = -dc1;
  float pos_l = (cnt    > 0.f) ? acc[4] / fmaxf(cnt,    1.f) : 0.f;
  float mis_l = (acc[6] > 0.f) ? acc[5] / fmaxf(acc[6], 1.f) : 0.f;
  float neg_l = (acc[8] > 0.f) ? acc[7] / fmaxf(acc[8], 1.f) : 0.f;
  out[0] = ce + dice + 5.f * (pos_l + mis_l + neg_l);
}

extern "C" void kernel_launch(void* const* d_in, const int* in_sizes, int n_in,
                              void* d_out, int out_size, void* d_ws, size_t ws_size,
                              hipStream_t stream) {
  (void)in_sizes; (void)n_in; (void)out_size; (void)ws_size;
  const float* feat = (const float*)d_in[0];
  const float* nout = (const float*)d_in[1];
  const int*   tgt  = (const int*)d_in[2];
  float* out = (float*)d_out;
  char* ws = (char*)d_ws;
  float*         acc  = (float*)(ws + OFF_ACC);
  unsigned*      rs   = (unsigned*)(ws + OFF_RS);
  float*         part = (float*)(ws + OFF_PART);
  float*         sim  = (float*)(ws + OFF_SIM);
  unsigned char* posm = (unsigned char*)(ws + OFF_POSM);
  unsigned char* mA   = (unsigned char*)(ws + OFF_MA);
  unsigned char* mB   = (unsigned char*)(ws + OFF_MB);

  const int dilBlocks = (NVOX + 255) / 256;
  const int rowBlocks = NVOX / WW;

  init_ws<<<1, 256, 0, stream>>>(acc, rs);

  // CE + dice raw reductions, pos mask
  ce_dice_partial<<<1024, 256, 0, stream>>>(nout, tgt, posm, part);
  reduce_cols<<<1, 256, 0, stream>>>(part, 1024, 4, acc);          // acc[0..3]

  // std over positives (TDM-staged mask chunk)
  std_partial<<<NVOX / STD_CHUNK, 256, 0, stream>>>(feat, posm, part);
  reduce_cols<<<1, 256, 0, stream>>>(part, NVOX / STD_CHUNK, CC, acc + 16);
  finalize_std<<<1, 32, 0, stream>>>(acc);

  // sim (WMMA GEMV) + pos_loss partials
  sim_wmma<<<NVOX / (SIM_WAVES * SIM_TILES * 16), 256, 0, stream>>>(feat, posm, acc, sim, part);
  reduce_cols<<<1, 256, 0, stream>>>(part, NVOX / (SIM_WAVES * SIM_TILES * 16), 1, acc + 4);

  // mis = dil10(pos) & !pos  (separable 21-wide max along W,H,D)
  dilate1d_w<<<rowBlocks, WW, 0, stream>>>(posm, mA);
  dilate1d<<<dilBlocks, 256, 0, stream>>>(mA, mB, 1);
  dilate1d<<<dilBlocks, 256, 0, stream>>>(mB, mA, 2);
  masked_relu_reduce<<<1024, 256, 0, stream>>>(sim, mA, posm, part);
  reduce_cols<<<1, 256, 0, stream>>>(part, 1024, 2, acc + 5);      // acc[5],acc[6]

  // exact top-250 threshold among negatives via 4-level radix select
  for (int L = 0; L < 4; ++L) {
    topk_hist<<<512, 256, 0, stream>>>(sim, posm, rs, L);
    topk_select<<<1, 1, 0, stream>>>(rs, L);
  }
  hsm_build<<<1024, 256, 0, stream>>>(sim, posm, rs, mB);

  // fin = dil10(hsm) & !pos
  dilate1d_w<<<rowBlocks, WW, 0, stream>>>(mB, mA);
  dilate1d<<<dilBlocks, 256, 0, stream>>>(mA, mB, 1);
  dilate1d<<<dilBlocks, 256, 0, stream>>>(mB, mA, 2);
  masked_relu_reduce<<<1024, 256, 0, stream>>>(sim, mA, posm, part);
  reduce_cols<<<1, 256, 0, stream>>>(part, 1024, 2, acc + 7);      // acc[7],acc[8]

  final_combine<<<1, 1, 0, stream>>>(acc, out);
}